// LinearGaussianSystem_22454089023536
// MI455X (gfx1250) — compile-verified
//
#include <hip/hip_runtime.h>

// CDNA5 (gfx1250) wave32 WMMA fragment types for V_WMMA_F32_16X16X4_F32:
//   A: 16x4 f32  -> 2 VGPRs/lane ; B: 4x16 f32 -> 2 VGPRs/lane ; C/D: 16x16 f32 -> 8 VGPRs/lane
typedef __attribute__((ext_vector_type(2))) float v2f;
typedef __attribute__((ext_vector_type(8))) float v8f;

// One wave32 block per batch element.
// obs[B,16], Rmat[B,16,16] (SPD), Hm[16,32], bias[16], pmean[32], pprec[32,32]
// out_mean[B,32], out_cov[B,32,32]
__global__ __launch_bounds__(32) void kalman_posterior_kernel(
    const float* __restrict__ obs,
    const float* __restrict__ Rmat,
    const float* __restrict__ Hm,
    const float* __restrict__ bias,
    const float* __restrict__ pmean,
    const float* __restrict__ pprec,
    float* __restrict__ out_mean,
    float* __restrict__ out_cov)
{
    const int b    = blockIdx.x;
    const int t    = threadIdx.x;   // 0..31
    const int half = t >> 4;        // 0: lanes 0-15, 1: lanes 16-31
    const int lm   = t & 15;

    __shared__ float Rl[16][17];    // R, then its Cholesky L (lower)
    __shared__ float idg16[16];     // 1/diag(L)
    __shared__ float Xl[16][33];    // X = L^{-1} H
    __shared__ float Pl[32][33];    // P -> Cholesky M -> W = M^{-1}
    __shared__ float idg[32];       // 1/diag(M)
    __shared__ float rhs[32];
    __shared__ float vsh[32];

    // ---- 1) stage R (256 floats, 8 per lane) ------------------------------
    {
        const float* src = Rmat + (size_t)b * 256;
        const int r0 = t >> 1, c0 = (t & 1) * 8;
        #pragma unroll
        for (int j = 0; j < 8; ++j)
            Rl[r0][c0 + j] = src[r0 * 16 + c0 + j];
    }
    __syncthreads();

    // ---- 2) Cholesky R = L L^T (right-looking, lanes = rows) --------------
    for (int k = 0; k < 16; ++k) {
        const float d    = sqrtf(Rl[k][k]);
        const float invd = 1.0f / d;
        if (t == k) { Rl[k][k] = d; idg16[k] = invd; }
        if (t > k && t < 16) Rl[t][k] *= invd;
        __syncthreads();
        if (t > k && t < 16) {
            const float lik = Rl[t][k];
            for (int j = k + 1; j <= t; ++j)
                Rl[t][j] -= lik * Rl[j][k];
        }
        __syncthreads();
    }

    // ---- 3) z = L^{-1}(y - bias)  (uniform across lanes) ------------------
    float z[16];
    {
        const float* ob = obs + (size_t)b * 16;
        #pragma unroll
        for (int k = 0; k < 16; ++k) {
            float acc = ob[k] - bias[k];
            #pragma unroll
            for (int j = 0; j < k; ++j) acc -= Rl[k][j] * z[j];
            z[k] = acc * idg16[k];
        }
    }

    // ---- 4) X = L^{-1} H, lane t owns column t ----------------------------
    {
        float x[16];
        #pragma unroll
        for (int k = 0; k < 16; ++k) {
            float acc = Hm[k * 32 + t];
            #pragma unroll
            for (int j = 0; j < k; ++j) acc -= Rl[k][j] * x[j];
            x[k] = acc * idg16[k];
            Xl[k][t] = x[k];
        }
    }
    __syncthreads();

    // ---- 5) P = pprec + X^T X   via V_WMMA_F32_16X16X4_F32 ----------------
    // Tile (I,J): A[m][k] = X[k][16I+m], B[k][n] = X[k][16J+n] -> same frag.
    v8f c00 = {}, c01 = {}, c10 = {}, c11 = {};
    #pragma unroll
    for (int k0 = 0; k0 < 16; k0 += 4) {
        const int kb = k0 + 2 * half;
        v2f f0, f1;
        f0.x = Xl[kb][lm];       f0.y = Xl[kb + 1][lm];
        f1.x = Xl[kb][16 + lm];  f1.y = Xl[kb + 1][16 + lm];
        c00 = __builtin_amdgcn_wmma_f32_16x16x4_f32(false, f0, false, f0, (short)0, c00, false, false);
        c01 = __builtin_amdgcn_wmma_f32_16x16x4_f32(false, f0, false, f1, (short)0, c01, false, false);
        c10 = __builtin_amdgcn_wmma_f32_16x16x4_f32(false, f1, false, f0, (short)0, c10, false, false);
        c11 = __builtin_amdgcn_wmma_f32_16x16x4_f32(false, f1, false, f1, (short)0, c11, false, false);
    }
    #pragma unroll
    for (int r = 0; r < 8; ++r) {
        const int m0 = r + 8 * half;      // C/D layout: VGPR r, lanes16-31 -> M=r+8
        Pl[m0][lm]            = c00[r] + pprec[m0 * 32 + lm];
        Pl[m0][16 + lm]       = c01[r] + pprec[m0 * 32 + 16 + lm];
        Pl[16 + m0][lm]       = c10[r] + pprec[(16 + m0) * 32 + lm];
        Pl[16 + m0][16 + lm]  = c11[r] + pprec[(16 + m0) * 32 + 16 + lm];
    }
    __syncthreads();

    // ---- 6) Cholesky P = M M^T -------------------------------------------
    for (int k = 0; k < 32; ++k) {
        const float d    = sqrtf(Pl[k][k]);
        const float invd = 1.0f / d;
        if (t == k) { Pl[k][k] = d; idg[k] = invd; }
        if (t > k) Pl[t][k] *= invd;
        __syncthreads();
        const float lik = Pl[t][k];
        for (int j = k + 1; j < 32; ++j)
            if (j <= t) Pl[t][j] -= lik * Pl[j][k];
        __syncthreads();
    }

    // ---- 7) W = M^{-1}: lane t forward-solves M w = e_t -------------------
    float w[32];
    #pragma unroll
    for (int k = 0; k < 32; ++k) {
        float acc = (k == t) ? 1.0f : 0.0f;
        #pragma unroll
        for (int j = 0; j < k; ++j) acc -= Pl[k][j] * w[j];
        w[k] = acc * idg[k];
    }
    __syncthreads();
    #pragma unroll
    for (int k = 0; k < 32; ++k) Pl[k][t] = w[k];   // Pl now holds W

    // rhs = pprec @ pmean + X^T z  (component t)
    {
        float pt = 0.0f;
        #pragma unroll 8
        for (int j = 0; j < 32; ++j) pt += pprec[t * 32 + j] * pmean[j];
        float res = 0.0f;
        #pragma unroll
        for (int k = 0; k < 16; ++k) res += Xl[k][t] * z[k];
        rhs[t] = pt + res;
    }
    __syncthreads();

    // v = W * rhs (lane t computes row t), then mu = W^T v
    {
        float acc = 0.0f;
        #pragma unroll
        for (int j = 0; j < 32; ++j) acc += Pl[t][j] * rhs[j];
        vsh[t] = acc;
    }
    __syncthreads();
    {
        float mu = 0.0f;
        #pragma unroll
        for (int k = 0; k < 32; ++k) mu += w[k] * vsh[k];
        out_mean[(size_t)b * 32 + t] = mu;
    }

    // ---- 8) Sigma = W^T W  (K=32) via WMMA, store straight from C-layout --
    v8f s00 = {}, s01 = {}, s10 = {}, s11 = {};
    #pragma unroll
    for (int k0 = 0; k0 < 32; k0 += 4) {
        const int kb = k0 + 2 * half;
        v2f f0, f1;
        f0.x = Pl[kb][lm];       f0.y = Pl[kb + 1][lm];
        f1.x = Pl[kb][16 + lm];  f1.y = Pl[kb + 1][16 + lm];
        s00 = __builtin_amdgcn_wmma_f32_16x16x4_f32(false, f0, false, f0, (short)0, s00, false, false);
        s01 = __builtin_amdgcn_wmma_f32_16x16x4_f32(false, f0, false, f1, (short)0, s01, false, false);
        s10 = __builtin_amdgcn_wmma_f32_16x16x4_f32(false, f1, false, f0, (short)0, s10, false, false);
        s11 = __builtin_amdgcn_wmma_f32_16x16x4_f32(false, f1, false, f1, (short)0, s11, false, false);
    }
    float* covb = out_cov + (size_t)b * 1024;
    #pragma unroll
    for (int r = 0; r < 8; ++r) {
        const int m0 = r + 8 * half;
        covb[m0 * 32 + lm]             = s00[r];
        covb[m0 * 32 + 16 + lm]        = s01[r];
        covb[(16 + m0) * 32 + lm]      = s10[r];
        covb[(16 + m0) * 32 + 16 + lm] = s11[r];
    }
}

extern "C" void kernel_launch(void* const* d_in, const int* in_sizes, int n_in,
                              void* d_out, int out_size, void* d_ws, size_t ws_size,
                              hipStream_t stream) {
    (void)n_in; (void)out_size; (void)d_ws; (void)ws_size;
    const float* obs   = (const float*)d_in[0];   // [B,16]
    const float* Rmat  = (const float*)d_in[1];   // [B,16,16]
    const float* Hm    = (const float*)d_in[2];   // [16,32]
    const float* bias  = (const float*)d_in[3];   // [16]
    const float* pmean = (const float*)d_in[4];   // [32]
    const float* pprec = (const float*)d_in[5];   // [32,32]

    const int B = in_sizes[0] / 16;
    float* out_mean = (float*)d_out;                       // [B,32] first
    float* out_cov  = out_mean + (size_t)B * 32;           // then [B,32,32]

    kalman_posterior_kernel<<<dim3(B), dim3(32), 0, stream>>>(
        obs, Rmat, Hm, bias, pmean, pprec, out_mean, out_cov);
}